// GlobalAttentionPool1d_45535243272352
// MI455X (gfx1250) — compile-verified
//
#include <hip/hip_runtime.h>
#include <hip/hip_bf16.h>

// ---------------------------------------------------------------------------
// GlobalAttentionPool1d for MI455X (gfx1250, wave32)
//
// HBM-bound problem (128 MiB of x, ~134 MFLOP -> <1 FLOP/byte). Strategy:
//  * one pass over x with online (flash-style) softmax
//  * async global->LDS double-buffered streaming (ASYNCcnt)
//  * scores via v_wmma_f32_16x16x4_f32 (16 rows per wave, K accumulated in f32)
// ---------------------------------------------------------------------------

typedef __attribute__((ext_vector_type(2))) float v2f;
typedef __attribute__((ext_vector_type(8))) float v8f;

#define B_DIM 32
#define F_DIM 4096
#define D_DIM 256
#define P_PART 64                          // partitions per batch
#define F_PER_PART (F_DIM / P_PART)        // 64 rows per partition
#define TILE_ROWS 16
#define TILES_PER_PART (F_PER_PART / TILE_ROWS)  // 4 tiles
#define PART_STRIDE (2 + D_DIM)            // floats per partial record: m, s, acc[256]

#define NEG_HUGE (-3.402823466e38f)

__device__ __forceinline__ float readlane_f(float v, int lane) {
  return __int_as_float(__builtin_amdgcn_readlane(__float_as_int(v), lane));
}

// Generic pointers to LDS carry the LDS byte offset in their low 32 bits
// (ISA 10.2, LDS aperture).
__device__ __forceinline__ unsigned lds_offset(const void* p) {
  return (unsigned)(unsigned long long)p;
}

__global__ __launch_bounds__(32)
void gap_partial_kernel(const float* __restrict__ x,
                        const unsigned char* __restrict__ mask,
                        const float* __restrict__ query,
                        float* __restrict__ ws) {
  __shared__ __align__(16) float q_lds[D_DIM];
  __shared__ __align__(16) float tile[2][TILE_ROWS * D_DIM];  // 2 x 16KiB

  const int lane = threadIdx.x;  // 0..31 (wave32)
  const int b = blockIdx.x / P_PART;
  const int p = blockIdx.x % P_PART;
  const int f_base = p * F_PER_PART;

  // stage query into LDS (feeds WMMA B-operand reads)
  #pragma unroll
  for (int j = 0; j < 8; ++j) q_lds[lane + 32 * j] = query[lane + 32 * j];
  __syncthreads();

  const float* xb = x + ((size_t)b * F_DIM + f_base) * D_DIM;
  const unsigned char* mb = mask + (size_t)b * F_DIM + f_base;

  const unsigned lds_base0 = lds_offset(&tile[0][0]);
  const unsigned lds_base1 = lds_offset(&tile[1][0]);

  // Async-copy one contiguous 16KiB tile (16 rows x 1KiB) into LDS.
  // 32 lanes x 16B x 32 issues; tracked on ASYNCcnt.
  auto prefetch_tile = [&](int t, int buf) {
    const char* src = (const char*)(xb + (size_t)t * TILE_ROWS * D_DIM);
    unsigned base = buf ? lds_base1 : lds_base0;
    #pragma unroll
    for (int k = 0; k < 32; ++k) {
      unsigned off = (unsigned)(k * 32 + lane) * 16u;
      asm volatile("global_load_async_to_lds_b128 %0, %1, %2"
                   :: "v"(base + off), "v"(off), "s"(src)
                   : "memory");
    }
  };

  float m = NEG_HUGE, s = 0.0f;
  float acc[8];
  #pragma unroll
  for (int j = 0; j < 8; ++j) acc[j] = 0.0f;

  prefetch_tile(0, 0);

  for (int t = 0; t < TILES_PER_PART; ++t) {
    const int buf = t & 1;
    if (t + 1 < TILES_PER_PART) {
      prefetch_tile(t + 1, (t + 1) & 1);
      // async loads complete in order: <=32 outstanding => tile t landed
      asm volatile("s_wait_asynccnt 32" ::: "memory");
    } else {
      asm volatile("s_wait_asynccnt 0" ::: "memory");
    }

    // ---- scores for 16 rows via v_wmma_f32_16x16x4_f32 ----
    // A layout (16x4 f32): lanes 0-15 hold K=0,1 in v0,v1; lanes 16-31 K=2,3.
    // B = query chunk broadcast across all 16 columns -> every column of D
    // holds the 16 scores; read column 0 at lanes 0 (M=0..7) / 16 (M=8..15).
    v8f c = {};
    const float* trow = &tile[buf][(lane & 15) * D_DIM];
    const int koff = (lane >> 4) * 2;  // 0 or 2
    #pragma unroll 8
    for (int d0 = 0; d0 < D_DIM; d0 += 4) {
      v2f a  = *(const v2f*)(trow + d0 + koff);
      v2f bq = *(const v2f*)(&q_lds[d0 + koff]);
      c = __builtin_amdgcn_wmma_f32_16x16x4_f32(
          /*neg_a=*/false, a, /*neg_b=*/false, bq,
          /*c_mod=*/(short)0, c, /*reuse_a=*/false, /*reuse_b=*/false);
    }

    float sc[16];
    #pragma unroll
    for (int r = 0; r < 8; ++r) {
      sc[r]     = readlane_f(c[r], 0);   // M = r
      sc[r + 8] = readlane_f(c[r], 16);  // M = r + 8
    }

    // mask bits for the 16 rows -> wave-uniform scalars
    int mval = (lane < 16) ? (int)mb[t * TILE_ROWS + lane] : 0;
    int mk[16];
    #pragma unroll
    for (int i = 0; i < 16; ++i) mk[i] = __builtin_amdgcn_readlane(mval, i);

    float tmax = NEG_HUGE;
    #pragma unroll
    for (int i = 0; i < 16; ++i)
      if (!mk[i]) tmax = fmaxf(tmax, sc[i]);

    if (tmax > NEG_HUGE) {  // at least one unmasked row in this tile
      const float nm = fmaxf(m, tmax);
      const float scale = __expf(m - nm);  // first tile: exp(-huge) -> 0
      m = nm;
      s *= scale;
      #pragma unroll
      for (int j = 0; j < 8; ++j) acc[j] *= scale;

      const float* tb = &tile[buf][0];
      #pragma unroll
      for (int i = 0; i < 16; ++i) {
        const float e = mk[i] ? 0.0f : __expf(sc[i] - nm);  // wave-uniform
        s += e;
        if (e != 0.0f) {
          #pragma unroll
          for (int j = 0; j < 8; ++j)   // lane owns d = lane + 32*j (bank-conflict-free)
            acc[j] = fmaf(e, tb[i * D_DIM + lane + 32 * j], acc[j]);
        }
      }
    }
  }

  // write partial record: [m, s, acc(256)]
  float* part = ws + (size_t)blockIdx.x * PART_STRIDE;
  if (lane == 0) { part[0] = m; part[1] = s; }
  #pragma unroll
  for (int j = 0; j < 8; ++j) part[2 + lane + 32 * j] = acc[j];
}

__global__ __launch_bounds__(256)
void gap_combine_kernel(const float* __restrict__ ws, float* __restrict__ out) {
  const int b = blockIdx.x;
  const int d = threadIdx.x;
  const float* base = ws + (size_t)b * P_PART * PART_STRIDE;

  float M = NEG_HUGE;
  for (int p = 0; p < P_PART; ++p) M = fmaxf(M, base[(size_t)p * PART_STRIDE]);

  float S = 0.0f, A = 0.0f;
  for (int p = 0; p < P_PART; ++p) {
    const float* pp = base + (size_t)p * PART_STRIDE;
    const float w = __expf(pp[0] - M);  // empty partition: m=-huge, s=0 -> no contribution
    S = fmaf(pp[1], w, S);
    A = fmaf(pp[2 + d], w, A);
  }
  // all-masked batch => S==0 => reference produces 0
  out[(size_t)b * D_DIM + d] = (S > 0.0f) ? (A / S) : 0.0f;
}

extern "C" void kernel_launch(void* const* d_in, const int* in_sizes, int n_in,
                              void* d_out, int out_size, void* d_ws, size_t ws_size,
                              hipStream_t stream) {
  const float* x = (const float*)d_in[0];
  const unsigned char* mask = (const unsigned char*)d_in[1];  // jax bool -> 1 byte
  const float* query = (const float*)d_in[2];
  float* out = (float*)d_out;
  float* ws = (float*)d_ws;  // needs P_PART*B_DIM*PART_STRIDE floats (~2.1 MB)

  gap_partial_kernel<<<B_DIM * P_PART, 32, 0, stream>>>(x, mask, query, ws);
  gap_combine_kernel<<<B_DIM, D_DIM, 0, stream>>>(ws, out);
}